// SimpleDrugEncoder_1812476199508
// MI455X (gfx1250) — compile-verified
//
#include <hip/hip_runtime.h>

#define N_NODES  100000
#define N_EDGES  3200000
#define N_GRAPHS 1024
#define IN_DIM   78
#define HID      128

typedef __attribute__((ext_vector_type(2))) float v2f;
typedef __attribute__((ext_vector_type(8))) float v8f;

// ---------------------------------------------------------------- utilities

__global__ void zero_f32_kernel(float* __restrict__ p, int n) {
    int i = blockIdx.x * blockDim.x + threadIdx.x;
    if (i < n) p[i] = 0.0f;
}

// ------------------------------------------------- edge scatter aggregation
// agg[dst[e]][f] += x[src[e]][f]  for all 3.2M edges, f in [0,78)
// One thread per (edge, feature): consecutive threads cover consecutive
// features of one edge -> coalesced gather, GLOBAL_ATOMIC_ADD_F32 scatter.
__global__ void edge_agg_kernel(const float* __restrict__ x,
                                const int*   __restrict__ edge_index,
                                float*       __restrict__ agg) {
    long long idx = (long long)blockIdx.x * blockDim.x + threadIdx.x;
    if (idx >= (long long)N_EDGES * IN_DIM) return;
    int e = (int)(idx / IN_DIM);
    int f = (int)(idx - (long long)e * IN_DIM);
    int s = edge_index[e];            // row 0 of [2, N_EDGES]
    int d = edge_index[N_EDGES + e];  // row 1
    unsafeAtomicAdd(&agg[d * IN_DIM + f], x[s * IN_DIM + f]);
}

// --------------------------------------- GEMM1: h1 = relu((x+agg)@W1 + b1)
// One wave computes one 16x16 output tile with V_WMMA_F32_16X16X4_F32.
// Block = 256 threads = 8 waves: blockIdx.x -> M tile, wave id -> N tile.
__global__ __launch_bounds__(256) void gin_mlp1_kernel(
    const float* __restrict__ x, const float* __restrict__ agg,
    const float* __restrict__ W1, const float* __restrict__ b1,
    float* __restrict__ h1) {
    const int wave = threadIdx.x >> 5;   // N tile 0..7
    const int lane = threadIdx.x & 31;
    const int g    = lane >> 4;          // half-wave group
    const int mn   = lane & 15;
    const int rowA = blockIdx.x * 16 + mn;  // A row loaded by this lane
    const int colB = wave * 16 + mn;        // B col loaded by this lane

    v8f c = {};
#pragma unroll
    for (int k0 = 0; k0 < 80; k0 += 4) {   // K=78 padded to 80
        v2f a, b;
#pragma unroll
        for (int v = 0; v < 2; ++v) {
            int k = k0 + g * 2 + v;
            bool ok = (k < IN_DIM);
            a[v] = ok ? (x[rowA * IN_DIM + k] + agg[rowA * IN_DIM + k]) : 0.0f;
            b[v] = ok ? W1[k * HID + colB] : 0.0f;
        }
        c = __builtin_amdgcn_wmma_f32_16x16x4_f32(false, a, false, b,
                                                  (short)0, c, false, false);
    }

    const float bias = b1[colB];
#pragma unroll
    for (int v = 0; v < 8; ++v) {
        int m = blockIdx.x * 16 + g * 8 + v;   // C layout: M = (lane/16)*8 + v
        h1[m * HID + colB] = fmaxf(c[v] + bias, 0.0f);
    }
}

// ------------------- GEMM2 fused with pool: sums[batch[m]] += h1@W2 + b2
__global__ __launch_bounds__(256) void mlp2_pool_kernel(
    const float* __restrict__ h1, const float* __restrict__ W2,
    const float* __restrict__ b2, const int* __restrict__ batch,
    float* __restrict__ sums) {
    const int wave = threadIdx.x >> 5;
    const int lane = threadIdx.x & 31;
    const int g    = lane >> 4;
    const int mn   = lane & 15;
    const int rowA = blockIdx.x * 16 + mn;
    const int colB = wave * 16 + mn;

    v8f c = {};
#pragma unroll
    for (int k0 = 0; k0 < HID; k0 += 4) {
        v2f a, b;
#pragma unroll
        for (int v = 0; v < 2; ++v) {
            int k = k0 + g * 2 + v;
            a[v] = h1[rowA * HID + k];
            b[v] = W2[k * HID + colB];
        }
        c = __builtin_amdgcn_wmma_f32_16x16x4_f32(false, a, false, b,
                                                  (short)0, c, false, false);
    }

    const float bias = b2[colB];
#pragma unroll
    for (int v = 0; v < 8; ++v) {
        int m = blockIdx.x * 16 + g * 8 + v;
        int gid = batch[m];                        // sorted -> L2-hot
        unsafeAtomicAdd(&sums[gid * HID + colB], c[v] + bias);
    }
}

// ------------------------------------------------------------ pool helpers
__global__ void count_kernel(const int* __restrict__ batch,
                             float* __restrict__ counts) {
    int i = blockIdx.x * blockDim.x + threadIdx.x;
    if (i < N_NODES) unsafeAtomicAdd(&counts[batch[i]], 1.0f);
}

__global__ void finalize_kernel(const float* __restrict__ sums,
                                const float* __restrict__ counts,
                                float* __restrict__ out) {
    int i = blockIdx.x * blockDim.x + threadIdx.x;
    if (i < N_GRAPHS * HID) out[i] = sums[i] / fmaxf(counts[i / HID], 1.0f);
}

// ----------------------------------------------------------------- launch
extern "C" void kernel_launch(void* const* d_in, const int* in_sizes, int n_in,
                              void* d_out, int out_size, void* d_ws, size_t ws_size,
                              hipStream_t stream) {
    const float* x          = (const float*)d_in[0];
    const int*   edge_index = (const int*)d_in[1];
    const int*   batch      = (const int*)d_in[2];
    const float* W1         = (const float*)d_in[3];
    const float* b1         = (const float*)d_in[4];
    const float* W2         = (const float*)d_in[5];
    const float* b2         = (const float*)d_in[6];
    float*       out        = (float*)d_out;

    // workspace layout: agg | h1 | sums | counts  (~83 MB)
    float* agg    = (float*)d_ws;
    float* h1     = agg  + (size_t)N_NODES * IN_DIM;
    float* sums   = h1   + (size_t)N_NODES * HID;
    float* counts = sums + (size_t)N_GRAPHS * HID;

    const int nAgg  = N_NODES * IN_DIM;
    const int nPool = N_GRAPHS * HID + N_GRAPHS;  // sums + counts contiguous
    zero_f32_kernel<<<(nAgg + 255) / 256, 256, 0, stream>>>(agg, nAgg);
    zero_f32_kernel<<<(nPool + 255) / 256, 256, 0, stream>>>(sums, nPool);

    const long long edgeWork = (long long)N_EDGES * IN_DIM;
    edge_agg_kernel<<<(int)((edgeWork + 255) / 256), 256, 0, stream>>>(
        x, edge_index, agg);

    gin_mlp1_kernel<<<N_NODES / 16, 256, 0, stream>>>(x, agg, W1, b1, h1);

    count_kernel<<<(N_NODES + 255) / 256, 256, 0, stream>>>(batch, counts);

    mlp2_pool_kernel<<<N_NODES / 16, 256, 0, stream>>>(h1, W2, b2, batch, sums);

    finalize_kernel<<<(N_GRAPHS * HID + 255) / 256, 256, 0, stream>>>(
        sums, counts, out);
}